// LightGCN_26663156974325
// MI455X (gfx1250) — compile-verified
//
#include <hip/hip_runtime.h>
#include <hip/hip_bf16.h>
#include <math.h>

#define NNODES 250000
#define DIMK   64
#define NE_    4000000
#define BB     4096
#define KK     40
#define REGC   1e-4f
#define TS     68                    // LDS row stride (floats): 272B rows, 16B aligned
#define PER_WAVE (3 * 16 * TS + 32)  // zp tile, zn tile, scratch tile, wp[16], wn[16]

typedef __attribute__((ext_vector_type(2))) float v2f;
typedef __attribute__((ext_vector_type(8))) float v8f;

#if __has_builtin(__builtin_amdgcn_wmma_f32_16x16x4_f32)
#define HAVE_WMMA 1
#endif

// D(16x16,f32) += A(16x4,f32) x B(4x16,f32) over K=64.
// A-frag layout (ISA 7.12.2): lanes 0-15 hold K=k0,k0+1 in v0,v1; lanes 16-31 K=k0+2,k0+3.
// B = W^T, W row-major [64 out][64 in]: B[k][n] = W[n][k] -> same float2 addressing as A.
__device__ __forceinline__ v8f wmma16x16(const float* A, int lda, const float* W,
                                         int n0, int m, int half, v8f c) {
#pragma unroll
  for (int k0 = 0; k0 < DIMK; k0 += 4) {
    v2f a = *(const v2f*)(A + m * lda + k0 + 2 * half);
    v2f b = *(const v2f*)(W + (n0 + m) * DIMK + k0 + 2 * half);
#ifdef HAVE_WMMA
    c = __builtin_amdgcn_wmma_f32_16x16x4_f32(false, a, false, b, (short)0, c,
                                              false, false);
#else
    c[0] += a.x * b.x;  // placeholder so the file still compiles if builtin absent
#endif
  }
  return c;
}

// out(16x64) = act(A(16x64) @ W^T + bias). ACT: 0=relu, 1=tanh.
// C/D layout: VGPR i holds row M=i+8*half, column N = (lane&15)+n0.
template <int ACT>
__device__ __forceinline__ void gemm16x64(const float* A, int lda, const float* W,
                                          const float* bias, float* out, int os,
                                          int lane) {
  int half = lane >> 4, m = lane & 15;
#pragma unroll
  for (int n0 = 0; n0 < DIMK; n0 += 16) {
    float bv = bias[n0 + m];
    v8f c;
#pragma unroll
    for (int i = 0; i < 8; ++i) c[i] = bv;
    c = wmma16x16(A, lda, W, n0, m, half, c);
#pragma unroll
    for (int i = 0; i < 8; ++i) {
      float x = c[i];
      if (ACT == 0) x = fmaxf(x, 0.f);
      if (ACT == 1) x = tanhf(x);
      out[(i + 8 * half) * os + n0 + m] = x;
    }
  }
}

__device__ __forceinline__ void wait_ds0() {
  asm volatile("s_wait_dscnt 0" ::: "memory");
}

// ---------------- graph branch ----------------

__global__ void k_degree(const int* __restrict__ src, float* __restrict__ deg, int ne) {
  int e = blockIdx.x * blockDim.x + threadIdx.x;
  if (e < ne) atomicAdd(&deg[src[e]], 1.0f);
}

__global__ void k_dinv(const float* __restrict__ deg, float* __restrict__ dinv, int nn) {
  int i = blockIdx.x * blockDim.x + threadIdx.x;
  if (i < nn) {
    float d = deg[i];
    dinv[i] = d > 0.f ? rsqrtf(d) : 0.f;
  }
}

// 16 threads per edge, one float4 chunk each: y[dst] += norm * x[src]
__global__ void k_conv(const float* __restrict__ x, const float* __restrict__ dinv,
                       const int* __restrict__ src, const int* __restrict__ dst,
                       float* __restrict__ y) {
  unsigned gid = blockIdx.x * blockDim.x + threadIdx.x;
  unsigned e = gid >> 4;
  if (e >= NE_) return;
  int c = gid & 15;
  int s = src[e], t = dst[e];
  float nrm = dinv[s] * dinv[t];
  float4 m = ((const float4*)(x + (size_t)s * DIMK))[c];
  float* yt = y + (size_t)t * DIMK + c * 4;
  atomicAdd(yt + 0, m.x * nrm);
  atomicAdd(yt + 1, m.y * nrm);
  atomicAdd(yt + 2, m.z * nrm);
  atomicAdd(yt + 3, m.w * nrm);
}

// ---------------- fused dense branch (WMMA) ----------------
// one wave per 16-row tile; 4 waves per block
__global__ void k_fused(const float* __restrict__ E1, const float* __restrict__ x1,
                        const float* __restrict__ x2, const float* __restrict__ E2,
                        const float* __restrict__ W0, const float* __restrict__ b0,
                        const float* __restrict__ W1, const float* __restrict__ b1,
                        const float* __restrict__ Wa, const float* __restrict__ ba,
                        const float* __restrict__ Wq, float* __restrict__ Z) {
  __shared__ float sm[4 * PER_WAVE];
  int lane = threadIdx.x & 31;
  int wave = threadIdx.x >> 5;
  int tile = blockIdx.x * 4 + wave;
  const int ntiles = NNODES / 16;
  if (tile >= ntiles) return;

  float* zp = sm + wave * PER_WAVE;
  float* zn = zp + 16 * TS;
  float* hs = zn + 16 * TS;   // scratch: h1, then tanh tiles
  float* wps = hs + 16 * TS;  // 16
  float* wns = wps + 16;      // 16

  size_t r0 = (size_t)tile * 16;
  const float* e1t = E1 + r0 * DIMK;
  const float* x1t = x1 + r0 * DIMK;
  const float* x2t = x2 + r0 * DIMK;

  // stage A: z_p = (E1 + x1 + x2)/3 -> LDS
  const float third = 1.0f / 3.0f;
#pragma unroll
  for (int i = 0; i < 8; ++i) {
    int idx = lane + 32 * i;      // 256 float4 chunks
    int m = idx >> 4, c4 = idx & 15;
    float4 a = ((const float4*)(e1t + m * DIMK))[c4];
    float4 b = ((const float4*)(x1t + m * DIMK))[c4];
    float4 c = ((const float4*)(x2t + m * DIMK))[c4];
    float4 z = make_float4((a.x + b.x + c.x) * third, (a.y + b.y + c.y) * third,
                           (a.z + b.z + c.z) * third, (a.w + b.w + c.w) * third);
    *(float4*)&zp[m * TS + c4 * 4] = z;
  }
  wait_ds0();

  // stage B: h = relu(E2 @ W0^T + b0), A from global
  gemm16x64<0>(E2 + r0 * DIMK, DIMK, W0, b0, hs, TS, lane);
  wait_ds0();

  // stage C: z_n = relu(h @ W1^T + b1), A from LDS
  gemm16x64<0>(hs, TS, W1, b1, zn, TS, lane);
  wait_ds0();

  // stage D: t = tanh(z_p @ Wa^T + ba); w_p = t @ Wq
  gemm16x64<1>(zp, TS, Wa, ba, hs, TS, lane);
  wait_ds0();
  if (lane < 16) {
    float s = 0.f;
#pragma unroll
    for (int k = 0; k < DIMK; ++k) s += hs[lane * TS + k] * Wq[k];
    wps[lane] = s;
  }
  wait_ds0();

  // stage E: t = tanh(z_n @ Wa^T + ba); w_n = t @ Wq
  gemm16x64<1>(zn, TS, Wa, ba, hs, TS, lane);
  wait_ds0();
  if (lane < 16) {
    float s = 0.f;
#pragma unroll
    for (int k = 0; k < DIMK; ++k) s += hs[lane * TS + k] * Wq[k];
    wns[lane] = s;
  }
  wait_ds0();

  // stage F: alpha = softmax([w_p, w_n]); Z = a0*z_p + a1*z_n -> global
#pragma unroll
  for (int i = 0; i < 8; ++i) {
    int idx = lane + 32 * i;
    int m = idx >> 4, c4 = idx & 15;
    float wpv = wps[m], wnv = wns[m];
    float mx = fmaxf(wpv, wnv);
    float e0 = __expf(wpv - mx), e1 = __expf(wnv - mx);
    float a0 = e0 / (e0 + e1), a1 = 1.0f - a0;
    float4 p = *(const float4*)&zp[m * TS + c4 * 4];
    float4 q = *(const float4*)&zn[m * TS + c4 * 4];
    float4 z = make_float4(a0 * p.x + a1 * q.x, a0 * p.y + a1 * q.y,
                           a0 * p.z + a1 * q.z, a0 * p.w + a1 * q.w);
    ((float4*)(Z + (r0 + m) * DIMK))[c4] = z;
  }
}

// ---------------- BPR loss ----------------

__device__ __forceinline__ float wred(float x) {
#pragma unroll
  for (int s = 16; s >= 1; s >>= 1) x += __shfl_xor(x, s, 32);
  return x;
}

__device__ __forceinline__ float log_sigmoid(float x) {
  return fminf(x, 0.f) - log1pf(__expf(-fabsf(x)));
}

__global__ void k_bpr(const float* __restrict__ Z, const int* __restrict__ uu,
                      const int* __restrict__ vv, const int* __restrict__ nn,
                      const float* __restrict__ w, float* __restrict__ out) {
  int lane = threadIdx.x & 31;
  int b = (blockIdx.x * blockDim.x + threadIdx.x) >> 5;
  if (b >= BB) return;
  const float* ur = Z + (size_t)uu[b] * DIMK;
  const float* vr = Z + (size_t)vv[b] * DIMK;
  float u0 = ur[lane], u1 = ur[lane + 32];
  float v0 = vr[lane], v1 = vr[lane + 32];
  float pos = wred(u0 * v0 + u1 * v1);
  float ww = w[b];
  float coef = ww > 0.f ? 1.0f : (ww < 0.f ? 2.0f : 1.5f);  // -0.5*sign+1.5
  float reg = u0 * u0 + u1 * u1 + v0 * v0 + v1 * v1;
  float bpr = 0.f;
  for (int k = 0; k < KK; ++k) {
    const float* nr = Z + (size_t)nn[b * KK + k] * DIMK;
    float n0 = nr[lane], n1 = nr[lane + 32];
    reg += n0 * n0 + n1 * n1;
    float neg = wred(u0 * n0 + u1 * n1);
    if (lane == 0) bpr += log_sigmoid(coef * pos - neg);
  }
  reg = wred(reg);
  if (lane == 0) atomicAdd(out, -bpr + REGC * reg);
}

extern "C" void kernel_launch(void* const* d_in, const int* in_sizes, int n_in,
                              void* d_out, int out_size, void* d_ws, size_t ws_size,
                              hipStream_t stream) {
  (void)in_sizes; (void)n_in; (void)ws_size;
  const float* E1 = (const float*)d_in[0];
  const float* E2 = (const float*)d_in[1];
  const float* W0 = (const float*)d_in[2];
  const float* b0 = (const float*)d_in[3];
  const float* W1 = (const float*)d_in[4];
  const float* b1 = (const float*)d_in[5];
  const float* Wa = (const float*)d_in[6];
  const float* ba = (const float*)d_in[7];
  const float* Wq = (const float*)d_in[8];
  const float* w  = (const float*)d_in[9];
  const int* ei   = (const int*)d_in[10];
  const int* uu   = (const int*)d_in[11];
  const int* vv   = (const int*)d_in[12];
  const int* nn   = (const int*)d_in[13];
  float* out = (float*)d_out;

  float* ws   = (float*)d_ws;
  float* deg  = ws;                        // NNODES
  float* dinv = deg + NNODES;              // NNODES
  float* x1   = dinv + NNODES;             // NNODES*64 (aliased as Z later)
  float* x2   = x1 + (size_t)NNODES * DIMK;// NNODES*64
  float* Z    = x1;                        // safe: fused kernel reads its x1 tile
                                           // before overwriting it with Z

  const int* src = ei;
  const int* dst = ei + NE_;

  hipMemsetAsync(deg, 0, NNODES * sizeof(float), stream);
  hipMemsetAsync(x1, 0, (size_t)NNODES * DIMK * sizeof(float), stream);
  hipMemsetAsync(x2, 0, (size_t)NNODES * DIMK * sizeof(float), stream);
  hipMemsetAsync(out, 0, (size_t)out_size * sizeof(float), stream);

  k_degree<<<(NE_ + 255) / 256, 256, 0, stream>>>(src, deg, NE_);
  k_dinv<<<(NNODES + 255) / 256, 256, 0, stream>>>(deg, dinv, NNODES);
  // x1 = conv(E1); x2 = conv(x1)
  k_conv<<<(NE_ * 16) / 256, 256, 0, stream>>>(E1, dinv, src, dst, x1);
  k_conv<<<(NE_ * 16) / 256, 256, 0, stream>>>(x1, dinv, src, dst, x2);

  const int ntiles = NNODES / 16;          // 15625
  k_fused<<<(ntiles + 3) / 4, 128, 0, stream>>>(E1, x1, x2, E2, W0, b0, W1, b1,
                                                Wa, ba, Wq, Z);

  k_bpr<<<(BB * 32) / 256, 256, 0, stream>>>(Z, uu, vv, nn, w, out);
}